// Wkv7_71038759076657
// MI455X (gfx1250) — compile-verified
//
#include <hip/hip_runtime.h>

// RWKV-7 scan for MI455X (gfx1250, wave32).
// One wave32 block per (head, 16-row strip): 32 heads x 4 strips = 128 blocks.
// State strip (16x64 f32) lives in 32 VGPRs in WMMA A-matrix layout.
// Per step: one chain of 16 V_WMMA_F32_16X16X4_F32 computes both
//   sa = S_old @ a          (B column 0)
//   y  = S_old @ (w*r)      (B column 1)
// and the output uses the exact identity
//   x = S_new @ r = y + sa*(b.r) + v*(k.r).
// B operand is built branchlessly: each lane reads its column's data through a
// per-lane LDS pointer (a | precomputed w*r | zeros), one ds_load_b64 per WMMA.
// No workgroup barriers: single-wave blocks rely on the in-order DS pipe.

typedef __attribute__((ext_vector_type(2))) float v2f;
typedef __attribute__((ext_vector_type(8))) float v8f;

#define NUM_HEADS 32
#define HEAD_SIZE 64
#define HXD (NUM_HEADS * HEAD_SIZE) /* 2048 */

__global__ __launch_bounds__(32)
void wkv7_scan_kernel(const float* __restrict__ gr, const float* __restrict__ gw,
                      const float* __restrict__ gk, const float* __restrict__ gv,
                      const float* __restrict__ ga, const float* __restrict__ gb,
                      const float* __restrict__ gs0,
                      float* __restrict__ xout, float* __restrict__ sout, int T)
{
    __shared__ __align__(16) float buf[2][6][HEAD_SIZE]; // r,w,k,v,a,b double-buffered
    __shared__ __align__(16) float lds_wr[HEAD_SIZE];    // w*r, rebuilt per step
    __shared__ __align__(16) float lds_zero[HEAD_SIZE];  // constant zeros
    __shared__ __align__(16) float lds_sa[16];           // sa broadcast scratch

    const int lane  = threadIdx.x;           // 0..31 (wave32)
    const int col   = lane & 15;             // N index (C/D layout) / M row (A layout)
    const int h     = lane >> 4;             // lane half (K sub-select in A/B layouts)
    const int head  = blockIdx.x >> 2;
    const int strip = blockIdx.x & 3;
    const int r0    = strip << 4;            // first row of this strip

    const float* vptr[6] = { gr, gw, gk, gv, ga, gb };

    // ---- load initial state strip into A-matrix layout registers ----
    // j(jb,p) = 16*jb + 4*(p>>1) + (p&1) + 2*h ; row = col.
    float S[4][8];
#pragma unroll
    for (int jb = 0; jb < 4; ++jb)
#pragma unroll
        for (int p = 0; p < 8; ++p) {
            const int j = 16*jb + 4*(p>>1) + (p&1) + 2*h;
            S[jb][p] = gs0[head*HEAD_SIZE*HEAD_SIZE + (r0+col)*HEAD_SIZE + j];
        }

    // ---- zero region + stage t = 0 ----
    lds_zero[lane]      = 0.f;
    lds_zero[lane + 32] = 0.f;
#pragma unroll
    for (int it = 0; it < 12; ++it) {
        const int vec = it >> 1;                     // compile-time after unroll
        const int d   = lane + ((it & 1) << 5);
        buf[0][vec][d] = vptr[vec][(size_t)head*HEAD_SIZE + d];
    }

    for (int t = 0; t < T; ++t) {
        const int cur = t & 1, nxt = cur ^ 1;

        // gfx1250 prefetch (global_prefetch_b8) a few steps ahead
        if (lane == 0 && t + 8 < T) {
            const size_t pb = (size_t)(t + 8) * HXD + (size_t)head * HEAD_SIZE;
            __builtin_prefetch(&gr[pb], 0, 1); __builtin_prefetch(&gr[pb+32], 0, 1);
            __builtin_prefetch(&gw[pb], 0, 1); __builtin_prefetch(&gw[pb+32], 0, 1);
            __builtin_prefetch(&gk[pb], 0, 1); __builtin_prefetch(&gk[pb+32], 0, 1);
            __builtin_prefetch(&gv[pb], 0, 1); __builtin_prefetch(&gv[pb+32], 0, 1);
            __builtin_prefetch(&ga[pb], 0, 1); __builtin_prefetch(&ga[pb+32], 0, 1);
            __builtin_prefetch(&gb[pb], 0, 1); __builtin_prefetch(&gb[pb+32], 0, 1);
        }

        // stage t+1 into the other buffer (same-wave DS is in-order)
        if (t + 1 < T) {
            const size_t tb = (size_t)(t + 1) * HXD + (size_t)head * HEAD_SIZE;
#pragma unroll
            for (int it = 0; it < 12; ++it) {
                const int vec = it >> 1;
                const int d   = lane + ((it & 1) << 5);
                buf[nxt][vec][d] = vptr[vec][tb + d];
            }
        }

        const float* lr = buf[cur][0];
        const float* lw = buf[cur][1];
        const float* lk = buf[cur][2];
        const float* lv = buf[cur][3];
        const float* la = buf[cur][4];
        const float* lb = buf[cur][5];

        // ---- build w*r column + dot products b.r, k.r ----
        const float rv0 = lr[lane],      rv1 = lr[lane + 32];
        const float wv0 = lw[lane],      wv1 = lw[lane + 32];
        const float bv0 = lb[lane],      bv1 = lb[lane + 32];
        const float kv0 = lk[lane],      kv1 = lk[lane + 32];
        lds_wr[lane]      = wv0 * rv0;
        lds_wr[lane + 32] = wv1 * rv1;
        float pbr = bv0 * rv0 + bv1 * rv1;
        float pkr = kv0 * rv0 + kv1 * rv1;
#pragma unroll
        for (int m = 16; m >= 1; m >>= 1) {
            pbr += __shfl_xor(pbr, m, 32);
            pkr += __shfl_xor(pkr, m, 32);
        }

        // per-lane branchless B source: col0 -> a, col1 -> w*r, else zeros
        const float* srcB = (col == 0) ? la : ((col == 1) ? lds_wr : lds_zero);

        // ---- WMMA chain: D = S_old(16x64) x [ a | w*r | 0 ... ] ----
        v8f acc = {0.f,0.f,0.f,0.f,0.f,0.f,0.f,0.f};
#pragma unroll
        for (int jb = 0; jb < 4; ++jb) {
#pragma unroll
            for (int c = 0; c < 4; ++c) {
                const int j0 = 16*jb + 4*c + 2*h;       // even -> 8B aligned
                const v2f Bv = *(const v2f*)(srcB + j0); // one ds_load_b64
                v2f Av;
                Av.x = S[jb][2*c];
                Av.y = S[jb][2*c + 1];
                acc = __builtin_amdgcn_wmma_f32_16x16x4_f32(
                          false, Av, false, Bv, (short)0, acc, false, false);
            }
        }

        // ---- scatter sa (D column 0 lives in lanes 0 and 16) ----
        if (col == 0) {
#pragma unroll
            for (int p = 0; p < 8; ++p) lds_sa[p + 8*h] = acc[p];
        }

        // ---- x = y + sa*(b.r) + v*(k.r); D column 1 lives in lanes 1 and 17 ----
        if (col == 1) {
            const size_t ob = (size_t)t * HXD + (size_t)head * HEAD_SIZE + r0 + 8*h;
#pragma unroll
            for (int p = 0; p < 8; ++p) {
                const float xv = acc[p] + lds_sa[p + 8*h] * pbr
                                        + lv[r0 + p + 8*h] * pkr;
                xout[ob + p] = xv;
            }
        }

        // ---- state update (elementwise in A-layout) ----
        const float sa_i = lds_sa[col];
        const float v_i  = lv[r0 + col];
#pragma unroll
        for (int jb = 0; jb < 4; ++jb) {
#pragma unroll
            for (int c = 0; c < 4; ++c) {
                const int j = 16*jb + 4*c + 2*h;
                S[jb][2*c]   = S[jb][2*c]  *lw[j]   + sa_i*lb[j]   + v_i*lk[j];
                S[jb][2*c+1] = S[jb][2*c+1]*lw[j+1] + sa_i*lb[j+1] + v_i*lk[j+1];
            }
        }
    }

    // ---- write final state ----
#pragma unroll
    for (int jb = 0; jb < 4; ++jb)
#pragma unroll
        for (int p = 0; p < 8; ++p) {
            const int j = 16*jb + 4*(p>>1) + (p&1) + 2*h;
            sout[head*HEAD_SIZE*HEAD_SIZE + (r0+col)*HEAD_SIZE + j] = S[jb][p];
        }
}

extern "C" void kernel_launch(void* const* d_in, const int* in_sizes, int n_in,
                              void* d_out, int out_size, void* d_ws, size_t ws_size,
                              hipStream_t stream) {
    (void)n_in; (void)d_ws; (void)ws_size; (void)out_size;
    // inputs: [0]=seq_length(int,1), [1]=r, [2]=w, [3]=k, [4]=v, [5]=a, [6]=b, [7]=state2
    const float* gr  = (const float*)d_in[1];
    const float* gw  = (const float*)d_in[2];
    const float* gk  = (const float*)d_in[3];
    const float* gv  = (const float*)d_in[4];
    const float* ga  = (const float*)d_in[5];
    const float* gb  = (const float*)d_in[6];
    const float* gs0 = (const float*)d_in[7];

    const int T = in_sizes[1] / HXD;          // r has T*H*D elements
    float* xout = (float*)d_out;              // [T, H, 1, D]
    float* sout = xout + (size_t)T * HXD;     // [H, D, D] appended

    wkv7_scan_kernel<<<dim3(NUM_HEADS * 4), dim3(32), 0, stream>>>(
        gr, gw, gk, gv, ga, gb, gs0, xout, sout, T);
}